// LocallyConnected2d_46591805227699
// MI455X (gfx1250) — compile-verified
//
#include <hip/hip_runtime.h>

// LocallyConnected2d: out[b,o,h,w] = sum_{c,k} patch[b,c,h,w,k] * W[o,c,h,w,k]
// Per (h,w): GEMM M=16(batch) N=64(out_c) K=576(c*9) via V_WMMA_F32_16X16X4_F32.
// Weight chunks (the bandwidth-dominant 462 MB stream) staged into LDS by the
// Tensor Data Mover via inline asm (portable across toolchains; bypasses the
// 5-arg vs 6-arg builtin arity difference), waited with s_wait_tensorcnt.

typedef __attribute__((ext_vector_type(2))) float v2f;
typedef __attribute__((ext_vector_type(8))) float v8f;
typedef __attribute__((ext_vector_type(4))) unsigned int u32x4;
typedef __attribute__((ext_vector_type(8))) int i32x8;
typedef __attribute__((ext_vector_type(4))) int i32x4;

#define B_ 16
#define C_ 64
#define H_ 56
#define W_ 56
#define O_ 64
#define WT 4            // consecutive w locations per block (weight runs of 4*9*4=144B)
#define CC 4            // c channels per K-chunk
#define KC (CC * 9)     // 36 K values per chunk -> 9 WMMA steps

// strides (in floats)
#define XS_B (C_ * H_ * W_)      // 200704
#define XS_C (H_ * W_)           // 3136
#define WS_O (C_ * H_ * W_ * 9)  // 1806336
#define WS_C (H_ * W_ * 9)       // 28224
#define OS_B (O_ * H_ * W_)      // 200704
#define OS_O (H_ * W_)           // 3136

// LDS layouts.
// B: padded-linear, element e=(o*CC+cc)*36 + (li*9+k) lives at dword e + e/4.
//    Matches TDM pad_enable (1 pad DWORD per 4 data DWORDs); o-stride becomes
//    180 dwords -> the 16 lanes of a B-operand read hit 16 distinct banks.
#define BS2_ELEMS (O_ * CC * (WT * 9))        // 9216 elements
#define BS2_SIZE (BS2_ELEMS + BS2_ELEMS / 4)  // 11520 padded dwords (45 KB)
#define AS_ROWSTRIDE (KC + 2)                 // 38 (even -> b64 aligned, bank stagger)
#define AS_SIZE (WT * B_ * AS_ROWSTRIDE)      // 2432 floats
#define LDS_FLOATS (BS2_SIZE + AS_SIZE)       // 13952 floats = 55808 B

__global__ __launch_bounds__(512) void lc2d_wmma_f32_kernel(
    const float* __restrict__ x, const float* __restrict__ wgt, float* __restrict__ out) {
  __shared__ float lds[LDS_FLOATS];
  float* Bs = lds;             // padded-linear B chunk
  float* As = lds + BS2_SIZE;  // As[(li*16+b)*38 + kidx]

  const int bid = blockIdx.x;
  const int h = bid / (W_ / WT);
  const int w0 = (bid % (W_ / WT)) * WT;

  const int tid = threadIdx.x;
  const int lane = tid & 31;
  const int wid = tid >> 5;            // 0..15 (uniform per wave)
  const int li = wid & (WT - 1);       // strip location for this wave
  const int n0 = (wid >> 2) * 16;      // o-tile base for this wave

  const int nlow = lane & 15;          // M (A) / N (B) index within tile
  const int khalf = (lane >> 4) << 1;  // lanes 16-31 hold K+2/K+3 halves
  const int li9 = li * 9;
  const int o4 = (n0 + nlow) * CC;     // o * CC for B element addressing

  v8f acc = {};

  const long wlocbase = (long)(h * W_ + w0) * 9;

#if defined(__gfx1250__)
  // LDS byte offset of Bs: low 32 bits of the flat LDS-aperture address.
  const unsigned int ldsB = (unsigned int)(unsigned long long)(uintptr_t)&lds[0];
  // D# group1: wg_mask=0 | data_size=4B | pad_enable | pad_interval=4DW | pad_amount=1DW;
  //            tensor_dim0=36, tensor_dim1=4, tile_dim0=36, tile_dim1=4, tile_dim2=64,
  //            tensor_dim0_stride=28224, tensor_dim1_stride=1806336 (0x1B9000).
  const i32x8 g1 = {
      (int)((2u << 16) | (1u << 20) | (1u << 22)),  // data_size=2, pad_en, pad_interval=1
      (int)(36u << 16),                             // tensor_dim0[15:0]
      (int)(4u << 16),                              // tensor_dim0[31:16]=0 | tensor_dim1[15:0]
      (int)(36u << 16),                             // tensor_dim1[31:16]=0 | tile_dim0
      (int)(4u | (64u << 16)),                      // tile_dim1 | tile_dim2
      (int)28224,                                   // tensor_dim0_stride[31:0]
      (int)0x90000000u,                             // s0[47:32]=0 | s1[15:0]=0x9000
      (int)0x1B                                     // tensor_dim1_stride[47:16]
  };
  const i32x4 g2 = {64, 0, 0, 0};  // tensor_dim2=64; tile_dim3=0
  const i32x4 g3 = {0, 0, 0, 0};
#endif

  for (int c0 = 0; c0 < C_; c0 += CC) {
    __syncthreads();  // previous chunk fully consumed before overwrite

#if defined(__gfx1250__)
    // ---- Stage B chunk with the Tensor Data Mover (one issue from wave 0).
    if (wid == 0) {
      unsigned long long ga = (unsigned long long)(uintptr_t)wgt +
                              4ull * ((unsigned long long)c0 * WS_C + (unsigned long long)wlocbase);
      u32x4 g0 = {1u,  // count=1, user mode
                  ldsB,
                  (unsigned int)(ga & 0xffffffffu),
                  (unsigned int)(((ga >> 32) & 0x1ffffffu) | (2u << 30))};  // type=2 ("image")
      asm volatile("tensor_load_to_lds %0, %1, %2, %3"
                   :
                   : "s"(g0), "s"(g1), "s"(g2), "s"(g3)
                   : "memory");
    }
#else
    // ---- Fallback (host pass / non-gfx1250): same padded-linear LDS layout.
    for (int e = tid; e < BS2_ELEMS; e += 512) {
      int o = e / (CC * WT * 9);
      int rr = e % (CC * WT * 9);  // cc*36 + r, r contiguous 36-float run
      int cc = rr / 36;
      int r = rr % 36;
      float val = wgt[(long)o * WS_O + (long)(c0 + cc) * WS_C + wlocbase + r];
      Bs[e + (e >> 2)] = val;
    }
#endif

    // ---- Stage A chunk: unfold 3x3 patches with zero padding (x is L2-resident).
    for (int e = tid; e < WT * B_ * KC; e += 512) {
      int lloc = e / (B_ * KC);
      int b = (e / KC) % B_;
      int kidx = e % KC;
      int cc = kidx / 9;
      int kq = kidx % 9;
      int hy = h + (kq / 3) - 1;
      int wx = w0 + lloc + (kq % 3) - 1;
      float val = 0.f;
      if (hy >= 0 && hy < H_ && wx >= 0 && wx < W_)
        val = x[(long)b * XS_B + (long)(c0 + cc) * XS_C + hy * W_ + wx];
      As[(lloc * B_ + b) * AS_ROWSTRIDE + kidx] = val;
    }

#if defined(__gfx1250__)
    if (wid == 0) __builtin_amdgcn_s_wait_tensorcnt(0);
#endif
    __syncthreads();

    // ---- 9 WMMA steps. A: one b64 LDS read; B: two b32 reads (padded layout).
    const float* arow = &As[(li * B_ + nlow) * AS_ROWSTRIDE];
#pragma unroll
    for (int s = 0; s < KC / 4; ++s) {
      int k0 = 4 * s + khalf;
      int k1 = k0 + 1;
      int cA = k0 / 9;  // K index -> (cc, tap); pairs can cross a c boundary
      int cB = k1 / 9;
      int e0 = (o4 + cA) * 36 + li9 + (k0 - cA * 9);
      int e1 = (o4 + cB) * 36 + li9 + (k1 - cB * 9);
      v2f a = *(const v2f*)(arow + k0);
      v2f b;
      b.x = Bs[e0 + (e0 >> 2)];
      b.y = Bs[e1 + (e1 >> 2)];
      acc = __builtin_amdgcn_wmma_f32_16x16x4_f32(
          /*neg_a=*/false, a, /*neg_b=*/false, b,
          /*c_mod=*/(short)0, acc, /*reuse_a=*/false, /*reuse_b=*/false);
    }
  }

  // ---- Stage D through LDS (reuse), then b128 stores.
  __syncthreads();
  {
    const int bbase = (lane >> 4) * 8;  // D layout: vgpr r -> M=r (+8 for upper lanes)
    const int o = n0 + nlow;
#pragma unroll
    for (int r = 0; r < 8; ++r) {
      int b = r + bbase;
      lds[((b * O_ + o) << 2) + li] = acc[r];  // Os[b][o][0..3] (4096 floats, fits)
    }
  }
  __syncthreads();
  for (int e = tid; e < B_ * O_; e += 512) {
    int b = e / O_;
    int o = e % O_;
    float4 v = *(const float4*)&lds[e << 2];
    *(float4*)&out[(long)b * OS_B + (long)o * OS_O + h * W_ + w0] = v;  // 16B aligned
  }
}

extern "C" void kernel_launch(void* const* d_in, const int* in_sizes, int n_in,
                              void* d_out, int out_size, void* d_ws, size_t ws_size,
                              hipStream_t stream) {
  const float* x = (const float*)d_in[0];  // [16,64,56,56] f32
  const float* w = (const float*)d_in[1];  // [1,64,64,56,56,9] f32
  float* out = (float*)d_out;              // [16,64,56,56] f32
  (void)in_sizes; (void)n_in; (void)out_size; (void)d_ws; (void)ws_size;

  dim3 grid(H_ * (W_ / WT));  // 56 * 14 = 784 blocks
  dim3 block(512);            // 16 wave32s: 4 strip locations x 4 o-tiles
  hipLaunchKernelGGL(lc2d_wmma_f32_kernel, grid, block, 0, stream, x, w, out);
}